// GCNEncoder_35253091566007
// MI455X (gfx1250) — compile-verified
//
#include <hip/hip_runtime.h>
#include <hip/hip_bf16.h>

#define NN   50000
#define INF  128
#define HID  64
#define LAT  32

typedef __attribute__((ext_vector_type(2))) float v2f;
typedef __attribute__((ext_vector_type(8))) float v8f;

// ---------------------------------------------------------------- utilities
__global__ void zero_f32(float* __restrict__ p, int n) {
    int i = blockIdx.x * blockDim.x + threadIdx.x;
    int stride = gridDim.x * blockDim.x;
    for (; i < n; i += stride) p[i] = 0.0f;
}

// ---------------------------------------------------------------- degrees
__global__ void degree_kernel(const int* __restrict__ src, const int* __restrict__ dst,
                              float* __restrict__ deg_out, float* __restrict__ deg_in, int E) {
    int e = blockIdx.x * blockDim.x + threadIdx.x;
    if (e < E) {
        atomicAdd(&deg_out[src[e]], 1.0f);
        atomicAdd(&deg_in[dst[e]], 1.0f);
    }
}

// deg -> deg^-0.5 in place (deg>=1 guaranteed by self loops; guard anyway)
__global__ void rsqrt_kernel(float* __restrict__ p, int n) {
    int i = blockIdx.x * blockDim.x + threadIdx.x;
    if (i < n) {
        float d = p[i];
        p[i] = (d > 0.0f) ? rsqrtf(fmaxf(d, 1.0f)) : 0.0f;
    }
}

// ---------------------------------------------------------------- GEMM 1 (WMMA f32)
// out[N,64] = (x[N,128] @ W[128,64]) * ns[:,None]
__global__ __launch_bounds__(256)
void gemm1_wmma(const float* __restrict__ x, const float* __restrict__ W,
                const float* __restrict__ ns, float* __restrict__ out) {
    const int lane = threadIdx.x & 31;
    const int wave = threadIdx.x >> 5;
    const int mTiles = NN / 16;          // 3125
    const int nTiles = HID / 16;         // 4
    int tile = blockIdx.x * (blockDim.x >> 5) + wave;   // wave-uniform
    if (tile >= mTiles * nTiles) return;                // uniform branch
    const int mt = tile / nTiles;
    const int nt = tile % nTiles;

    const int rowA  = mt * 16 + (lane & 15);            // A row (both lane halves)
    const int colB  = nt * 16 + (lane & 15);            // B col
    const int kHalf = (lane >> 4) * 2;                  // 0 or 2

    v8f acc = {};
    #pragma unroll
    for (int k = 0; k < INF; k += 4) {
        v2f a = *(const v2f*)(x + rowA * INF + k + kHalf);
        v2f b;
        b.x = W[(k + kHalf) * HID + colB];
        b.y = W[(k + kHalf + 1) * HID + colB];
        acc = __builtin_amdgcn_wmma_f32_16x16x4_f32(
            /*neg_a=*/false, a, /*neg_b=*/false, b,
            /*c_mod=*/(short)0, acc, /*reuse_a=*/false, /*reuse_b=*/false);
    }

    const int rBase = mt * 16 + 8 * (lane >> 4);
    const int c = nt * 16 + (lane & 15);
    #pragma unroll
    for (int j = 0; j < 8; ++j) {
        int r = rBase + j;
        out[r * HID + c] = acc[j] * ns[r];
    }
}

// ---------------------------------------------------------------- scatter layer 1
// acc[dst[e], c] += t[src[e], c]  for c in 0..63 ; one edge per 64 consecutive tids
__global__ void scatter64(const int* __restrict__ src, const int* __restrict__ dst,
                          const float* __restrict__ t, float* __restrict__ acc, int E) {
    unsigned tid = blockIdx.x * blockDim.x + threadIdx.x;
    unsigned e = tid >> 6;
    unsigned c = tid & 63u;
    if (e < (unsigned)E) {
        int s = src[e], d = dst[e];
        atomicAdd(&acc[(size_t)d * HID + c], t[(size_t)s * HID + c]);
    }
}

// h = relu(h * nd + b1)
__global__ void relu_bias(float* __restrict__ h, const float* __restrict__ nd,
                          const float* __restrict__ b1) {
    int i = blockIdx.x * blockDim.x + threadIdx.x;
    if (i < NN * HID) {
        int r = i >> 6, c = i & 63;
        h[i] = fmaxf(h[i] * nd[r] + b1[c], 0.0f);
    }
}

// ---------------------------------------------------------------- GEMM 2 (WMMA f32, both heads)
// tmu[N,32] = (h @ Wmu)*ns ; tls[N,32] = (h @ Wls)*ns   (K = 64)
__global__ __launch_bounds__(256)
void gemm2_wmma(const float* __restrict__ h, const float* __restrict__ Wmu,
                const float* __restrict__ Wls, const float* __restrict__ ns,
                float* __restrict__ tmu, float* __restrict__ tls) {
    const int lane = threadIdx.x & 31;
    const int wave = threadIdx.x >> 5;
    const int mTiles = NN / 16;          // 3125
    const int nTiles = LAT / 16;         // 2
    int tile = blockIdx.x * (blockDim.x >> 5) + wave;   // wave-uniform
    if (tile >= mTiles * nTiles * 2) return;
    const int mat = tile / (mTiles * nTiles);           // 0 = mu, 1 = ls
    const int rem = tile % (mTiles * nTiles);
    const int mt = rem / nTiles;
    const int nt = rem % nTiles;
    const float* __restrict__ W = mat ? Wls : Wmu;
    float* __restrict__ out = mat ? tls : tmu;

    const int rowA  = mt * 16 + (lane & 15);
    const int colB  = nt * 16 + (lane & 15);
    const int kHalf = (lane >> 4) * 2;

    v8f acc = {};
    #pragma unroll
    for (int k = 0; k < HID; k += 4) {
        v2f a = *(const v2f*)(h + rowA * HID + k + kHalf);
        v2f b;
        b.x = W[(k + kHalf) * LAT + colB];
        b.y = W[(k + kHalf + 1) * LAT + colB];
        acc = __builtin_amdgcn_wmma_f32_16x16x4_f32(
            false, a, false, b, (short)0, acc, false, false);
    }

    const int rBase = mt * 16 + 8 * (lane >> 4);
    const int c = nt * 16 + (lane & 15);
    #pragma unroll
    for (int j = 0; j < 8; ++j) {
        int r = rBase + j;
        out[r * LAT + c] = acc[j] * ns[r];
    }
}

// ---------------------------------------------------------------- scatter layer 2 (both heads)
// tid layout: 64 consecutive tids per edge; low 32 -> mu, high 32 -> ls.
// Each 32-thread wave handles one (edge, head) pair -> coalesced.
__global__ void scatter32x2(const int* __restrict__ src, const int* __restrict__ dst,
                            const float* __restrict__ tmu, const float* __restrict__ tls,
                            float* __restrict__ omu, float* __restrict__ ols, int E) {
    unsigned tid = blockIdx.x * blockDim.x + threadIdx.x;
    unsigned e = tid >> 6;
    unsigned r = tid & 63u;
    unsigned mat = r >> 5;
    unsigned c = r & 31u;
    if (e < (unsigned)E) {
        int s = src[e], d = dst[e];
        const float* __restrict__ t = mat ? tls : tmu;
        float* __restrict__ o = mat ? ols : omu;
        atomicAdd(&o[(size_t)d * LAT + c], t[(size_t)s * LAT + c]);
    }
}

// out = out * nd + bias   (applied to both halves of d_out)
__global__ void final_bias(float* __restrict__ omu, float* __restrict__ ols,
                           const float* __restrict__ nd, const float* __restrict__ bmu,
                           const float* __restrict__ bls) {
    int i = blockIdx.x * blockDim.x + threadIdx.x;
    int total = NN * LAT;
    if (i < 2 * total) {
        int mat = i >= total;
        int j = mat ? (i - total) : i;
        int r = j >> 5, c = j & 31;
        if (mat) ols[j] = ols[j] * nd[r] + bls[c];
        else     omu[j] = omu[j] * nd[r] + bmu[c];
    }
}

// ---------------------------------------------------------------- launch
extern "C" void kernel_launch(void* const* d_in, const int* in_sizes, int n_in,
                              void* d_out, int out_size, void* d_ws, size_t ws_size,
                              hipStream_t stream) {
    const float* x   = (const float*)d_in[0];
    const int*   src = (const int*)d_in[1];
    const int*   dst = (const int*)d_in[2];
    const float* W1  = (const float*)d_in[3];
    const float* b1  = (const float*)d_in[4];
    const float* Wmu = (const float*)d_in[5];
    const float* bmu = (const float*)d_in[6];
    const float* Wls = (const float*)d_in[7];
    const float* bls = (const float*)d_in[8];
    const int E = in_sizes[1];

    // workspace layout (floats)
    float* ws      = (float*)d_ws;
    float* deg_out = ws;                    // N   (becomes ns)
    float* deg_in  = ws + NN;               // N   (becomes nd)
    float* xw      = ws + 2 * NN;           // N*64
    float* h       = xw + (size_t)NN * HID; // N*64
    float* tmu     = h + (size_t)NN * HID;  // N*32
    float* tls     = tmu + (size_t)NN * LAT;// N*32

    float* omu = (float*)d_out;             // N*32
    float* ols = omu + (size_t)NN * LAT;    // N*32

    const int B = 256;

    // 1) zero accumulators (degrees, h, d_out)
    zero_f32<<<1024, B, 0, stream>>>(deg_out, 2 * NN);
    zero_f32<<<2048, B, 0, stream>>>(h, NN * HID);
    zero_f32<<<2048, B, 0, stream>>>(omu, 2 * NN * LAT);

    // 2) degrees, 3) rsqrt (in place -> ns, nd)
    degree_kernel<<<(E + B - 1) / B, B, 0, stream>>>(src, dst, deg_out, deg_in, E);
    rsqrt_kernel<<<(2 * NN + B - 1) / B, B, 0, stream>>>(deg_out, 2 * NN);

    // 4) xw = (x@W1)*ns  — WMMA, 16x16 tile per wave
    {
        int tiles = (NN / 16) * (HID / 16);         // 12500
        int wavesPerBlock = B / 32;
        gemm1_wmma<<<(tiles + wavesPerBlock - 1) / wavesPerBlock, B, 0, stream>>>(x, W1, deg_out, xw);
    }

    // 5) h[dst] += xw[src]
    scatter64<<<((unsigned)E * HID + B - 1) / B, B, 0, stream>>>(src, dst, xw, h, E);

    // 6) h = relu(h*nd + b1)
    relu_bias<<<(NN * HID + B - 1) / B, B, 0, stream>>>(h, deg_in, b1);

    // 7) tmu/tls = (h@W)*ns — WMMA
    {
        int tiles = (NN / 16) * (LAT / 16) * 2;     // 12500
        int wavesPerBlock = B / 32;
        gemm2_wmma<<<(tiles + wavesPerBlock - 1) / wavesPerBlock, B, 0, stream>>>(h, Wmu, Wls, deg_out, tmu, tls);
    }

    // 8) out[dst] += t[src] for both heads
    scatter32x2<<<((unsigned)E * 64 + B - 1) / B, B, 0, stream>>>(src, dst, tmu, tls, omu, ols, E);

    // 9) out = out*nd + bias
    final_bias<<<(2 * NN * LAT + B - 1) / B, B, 0, stream>>>(omu, ols, deg_in, bmu, bls);
}